// NodeModel_7138235646189
// MI455X (gfx1250) — compile-verified
//
#include <hip/hip_runtime.h>
#include <hip/hip_bf16.h>

typedef __attribute__((ext_vector_type(2))) float v2f;
typedef __attribute__((ext_vector_type(8))) float v8f;
typedef __attribute__((ext_vector_type(4))) int   v4i;

#define TM 32    // block tile rows
#define TN 128   // block tile cols (4 waves x 32 cols)
#define KC 32    // K chunk staged in LDS

#if __has_builtin(__builtin_amdgcn_global_load_async_to_lds_b128)
#define HAVE_ASYNC_LDS 1
// clang's builtin signature: (v4i addrspace(1)*, v4i addrspace(3)*, imm, imm)
typedef __attribute__((address_space(1))) v4i* gptr_b128;
typedef __attribute__((address_space(3))) v4i* lptr_b128;
#else
#define HAVE_ASYNC_LDS 0
#endif

// Fused GEMM: O[M x N] = act(A[M x K] @ B[K x N] + bias)
// MODE 0: A is a plain row-major matrix.
// MODE 1: A(m,k) = k<128 ? x[m][k] : k<384 ? mean[m][k-128] : u[batch[m]][k-384]
//         (fused concat for MLP2; K = 448; region is uniform per KC chunk)
// K must be a multiple of KC (128 / 256 / 448 all are).
template <int MODE, bool RELU>
__global__ __launch_bounds__(256) void gemm_wmma_f32(
    const float* __restrict__ A, const float* __restrict__ meanp,
    const float* __restrict__ up, const int* __restrict__ batch,
    const float* __restrict__ B, const float* __restrict__ bias,
    float* __restrict__ O, int M, int K, int N)
{
    // Row strides are multiples of 16B so b128 LDS writes stay aligned.
    __shared__ float As[2][TM][KC + 4];   // stride 36 floats = 144B
    __shared__ float Bs[2][KC][TN + 4];   // stride 132 floats = 528B

    const int tid  = threadIdx.x;
    const int lane = tid & 31;
    const int wave = tid >> 5;        // 8 waves
    const int wr   = wave & 1;        // 2 wave-rows (16 rows each)
    const int wc   = wave >> 1;       // 4 wave-cols (32 cols each)
    const int half = lane >> 4;       // frag: lanes 16-31 hold K+2,K+3
    const int l    = lane & 15;

    const int m0 = blockIdx.x * TM;
    const int n0 = blockIdx.y * TN;

    // ---- per-thread staging coordinates (fixed across K loop) ----
    // A tile: one b128 per thread (32 rows x 8 float4).
    const int rA  = tid >> 3;          // 0..31
    const int k4A = (tid & 7) * 4;     // float offset within chunk
    const int mA  = m0 + rA;
    const bool rowOK = (mA < M);
    // Clamp OOB rows to a valid address: A-tile row r only feeds output row
    // m0+r, and OOB rows are never stored, so stale data there is harmless.
    const int mc  = rowOK ? mA : (M - 1);

    const float* baseA0 = nullptr;     // MODE 0 row
    const float* px = nullptr;         // MODE 1 rows
    const float* pm = nullptr;
    const float* pu = nullptr;
    if (MODE == 0) {
        baseA0 = A + (size_t)mc * (unsigned)K;
    } else {
        px = A + (size_t)mc * 128u;
        pm = meanp + (size_t)mc * 256u;
        pu = up + (size_t)batch[mc] * 64u;
    }

    // B tile: four b128 per thread (32 rows x 32 float4).
    const int r0  = tid >> 5;          // 0..7 (rows r0, r0+8, r0+16, r0+24)
    const int c4  = (tid & 31) * 4;    // float col within tile

    // Resolve the A-source row pointer for a given chunk (uniform branch on kc0).
    auto a_src = [&](int kc0) -> const float* {
        if (MODE == 0) return baseA0 + kc0 + k4A;
        if (kc0 < 128) return px + kc0 + k4A;
        if (kc0 < 384) return pm + (kc0 - 128) + k4A;
        return pu + (kc0 - 384) + k4A;
    };

    v8f acc0 = {};
    v8f acc1 = {};

    const int nch = K / KC;

#if HAVE_ASYNC_LDS
    // ---- async direct global->LDS staging, double buffered (ASYNCcnt) ----
    auto issue = [&](int buf, int kc0) {
        const float* ap = a_src(kc0);
        const float* bb = B + (size_t)kc0 * (unsigned)N + n0 + c4;
        __builtin_amdgcn_global_load_async_to_lds_b128(
            (gptr_b128)ap, (lptr_b128)&As[buf][rA][k4A], 0, 0);
        __builtin_amdgcn_global_load_async_to_lds_b128(
            (gptr_b128)(bb + (size_t)(r0     ) * (unsigned)N),
            (lptr_b128)&Bs[buf][r0     ][c4], 0, 0);
        __builtin_amdgcn_global_load_async_to_lds_b128(
            (gptr_b128)(bb + (size_t)(r0 + 8 ) * (unsigned)N),
            (lptr_b128)&Bs[buf][r0 + 8 ][c4], 0, 0);
        __builtin_amdgcn_global_load_async_to_lds_b128(
            (gptr_b128)(bb + (size_t)(r0 + 16) * (unsigned)N),
            (lptr_b128)&Bs[buf][r0 + 16][c4], 0, 0);
        __builtin_amdgcn_global_load_async_to_lds_b128(
            (gptr_b128)(bb + (size_t)(r0 + 24) * (unsigned)N),
            (lptr_b128)&Bs[buf][r0 + 24][c4], 0, 0);
    };

    issue(0, 0);
    for (int ic = 0; ic < nch; ++ic) {
        const int cur = ic & 1;
        if (ic + 1 < nch) {
            // Prefetch next chunk (its 5 async ops may stay outstanding),
            // then wait until THIS chunk's 5 ops have landed (in-order).
            issue(cur ^ 1, (ic + 1) * KC);
            asm volatile("s_wait_asynccnt 0x5" ::: "memory");
        } else {
            asm volatile("s_wait_asynccnt 0x0" ::: "memory");
        }
        __syncthreads();   // all waves' async writes visible

        const int nc0 = wc * 32;
#pragma unroll
        for (int kk = 0; kk < KC; kk += 4) {
            v2f a, b0, b1;
            a.x  = As[cur][wr * 16 + l][kk + 2 * half];
            a.y  = As[cur][wr * 16 + l][kk + 2 * half + 1];
            b0.x = Bs[cur][kk + 2 * half][nc0 + l];
            b0.y = Bs[cur][kk + 2 * half + 1][nc0 + l];
            b1.x = Bs[cur][kk + 2 * half][nc0 + 16 + l];
            b1.y = Bs[cur][kk + 2 * half + 1][nc0 + 16 + l];
            acc0 = __builtin_amdgcn_wmma_f32_16x16x4_f32(
                false, a, false, b0, (short)0, acc0, false, false);
            acc1 = __builtin_amdgcn_wmma_f32_16x16x4_f32(
                false, a, false, b1, (short)0, acc1, false, false);
        }
        __syncthreads();   // safe to overwrite 'cur' buffer next iteration
    }
#else
    // ---- fallback: register staging with b128 loads/stores ----
    for (int ic = 0; ic < nch; ++ic) {
        const int kc0 = ic * KC;
        float4 av = *(const float4*)a_src(kc0);
        const float* baseB = B + (size_t)kc0 * (unsigned)N + n0 + c4;
        float4 bv0 = *(const float4*)(baseB + (size_t)(r0     ) * (unsigned)N);
        float4 bv1 = *(const float4*)(baseB + (size_t)(r0 + 8 ) * (unsigned)N);
        float4 bv2 = *(const float4*)(baseB + (size_t)(r0 + 16) * (unsigned)N);
        float4 bv3 = *(const float4*)(baseB + (size_t)(r0 + 24) * (unsigned)N);
        *(float4*)&As[0][rA][k4A]      = av;
        *(float4*)&Bs[0][r0     ][c4]  = bv0;
        *(float4*)&Bs[0][r0 + 8 ][c4]  = bv1;
        *(float4*)&Bs[0][r0 + 16][c4]  = bv2;
        *(float4*)&Bs[0][r0 + 24][c4]  = bv3;
        __syncthreads();

        const int nc0 = wc * 32;
#pragma unroll
        for (int kk = 0; kk < KC; kk += 4) {
            v2f a, b0, b1;
            a.x  = As[0][wr * 16 + l][kk + 2 * half];
            a.y  = As[0][wr * 16 + l][kk + 2 * half + 1];
            b0.x = Bs[0][kk + 2 * half][nc0 + l];
            b0.y = Bs[0][kk + 2 * half + 1][nc0 + l];
            b1.x = Bs[0][kk + 2 * half][nc0 + 16 + l];
            b1.y = Bs[0][kk + 2 * half + 1][nc0 + 16 + l];
            acc0 = __builtin_amdgcn_wmma_f32_16x16x4_f32(
                false, a, false, b0, (short)0, acc0, false, false);
            acc1 = __builtin_amdgcn_wmma_f32_16x16x4_f32(
                false, a, false, b1, (short)0, acc1, false, false);
        }
        __syncthreads();
    }
#endif

    // ---- epilogue: C/D layout — VGPR v holds row v (lanes 0-15) / v+8 (16-31) ----
    const int nA = n0 + wc * 32 + l;
    const int nB = nA + 16;
    const float biasA = bias[nA];
    const float biasB = bias[nB];
#pragma unroll
    for (int vv = 0; vv < 8; ++vv) {
        int m = m0 + wr * 16 + vv + 8 * half;
        if (m < M) {
            float vA = acc0[vv] + biasA;
            float vB = acc1[vv] + biasB;
            if (RELU) { vA = fmaxf(vA, 0.f); vB = fmaxf(vB, 0.f); }
            O[(size_t)m * (unsigned)N + nA] = vA;
            O[(size_t)m * (unsigned)N + nB] = vB;
        }
    }
}

// Zero scratch (sums + cnt) every call — graph-safe, deterministic.
__global__ void zero_kernel(float* p, size_t n)
{
    size_t i = (size_t)blockIdx.x * blockDim.x + threadIdx.x;
    size_t stride = (size_t)gridDim.x * blockDim.x;
    for (; i < n; i += stride) p[i] = 0.0f;
}

// Scatter-add h[row[e]] (256 floats) into sums[col[e]]; 64 threads/edge, float4 each.
__global__ void scatter_kernel(const float* __restrict__ h,
                               const int* __restrict__ row,
                               const int* __restrict__ col,
                               float* __restrict__ sums,
                               float* __restrict__ cnt, int E)
{
    int t  = blockIdx.x * blockDim.x + threadIdx.x;
    int e  = t >> 6;
    int f4 = t & 63;
    if (e >= E) return;
    int r = row[e];
    int c = col[e];
    float4 v = *((const float4*)(h + (size_t)r * 256) + f4);
    float* dst = sums + (size_t)c * 256 + f4 * 4;
    atomicAdd(dst + 0, v.x);
    atomicAdd(dst + 1, v.y);
    atomicAdd(dst + 2, v.z);
    atomicAdd(dst + 3, v.w);
    if (f4 == 0) atomicAdd(cnt + c, 1.0f);
}

// mean = sums / max(cnt, 1)  (in place)
__global__ void mean_kernel(float* __restrict__ sums, const float* __restrict__ cnt, int Nn)
{
    int t = blockIdx.x * blockDim.x + threadIdx.x;
    int node = t >> 6;
    int f4 = t & 63;
    if (node >= Nn) return;
    float inv = 1.0f / fmaxf(cnt[node], 1.0f);
    float4* p = (float4*)(sums + (size_t)node * 256) + f4;
    float4 v = *p;
    v.x *= inv; v.y *= inv; v.z *= inv; v.w *= inv;
    *p = v;
}

extern "C" void kernel_launch(void* const* d_in, const int* in_sizes, int n_in,
                              void* d_out, int out_size, void* d_ws, size_t ws_size,
                              hipStream_t stream)
{
    const float* x     = (const float*)d_in[0];
    const int*   ei    = (const int*)d_in[1];
    const float* u     = (const float*)d_in[2];
    const int*   batch = (const int*)d_in[3];
    const float* W1    = (const float*)d_in[4];
    const float* b1    = (const float*)d_in[5];
    const float* W2    = (const float*)d_in[6];
    const float* b2    = (const float*)d_in[7];
    const float* W3    = (const float*)d_in[8];
    const float* b3    = (const float*)d_in[9];
    const float* W4    = (const float*)d_in[10];
    const float* b4    = (const float*)d_in[11];
    float* out = (float*)d_out;

    const int F = 128, H = 256;
    const int M = in_sizes[0] / F;   // 50000 nodes
    const int E = in_sizes[1] / 2;   // 800000 edges

    // Workspace layout
    float* t0   = (float*)d_ws;                 // [M,H] MLP1 hidden, reused as MLP2 hidden
    float* h    = t0 + (size_t)M * H;           // [M,H] per-node MLP1 output
    float* sums = h + (size_t)M * H;            // [M,H] scatter sums -> mean
    float* cnt  = sums + (size_t)M * H;         // [M]   in-degree

    const int* row = ei;
    const int* col = ei + E;

    // 0) zero sums + cnt (contiguous)
    zero_kernel<<<2048, 256, 0, stream>>>(sums, (size_t)M * H + (size_t)M);

    const int mb = (M + TM - 1) / TM;

    // 1) t0 = relu(x @ W1 + b1)      [M,128]x[128,256]
    gemm_wmma_f32<0, true><<<dim3(mb, H / TN), 256, 0, stream>>>(
        x, nullptr, nullptr, nullptr, W1, b1, t0, M, F, H);

    // 2) h = t0 @ W2 + b2            [M,256]x[256,256]  (MLP1 per-NODE, not per-edge)
    gemm_wmma_f32<0, false><<<dim3(mb, H / TN), 256, 0, stream>>>(
        t0, nullptr, nullptr, nullptr, W2, b2, h, M, H, H);

    // 3) scatter-add h[row] -> sums[col], count in-degree
    int sthreads = E * 64;
    scatter_kernel<<<(sthreads + 255) / 256, 256, 0, stream>>>(h, row, col, sums, cnt, E);

    // 4) mean = sums / max(cnt,1)
    mean_kernel<<<(M * 64 + 255) / 256, 256, 0, stream>>>(sums, cnt, M);

    // 5) t0 = relu(concat(x, mean, u[batch]) @ W3 + b3)   K=448, concat fused in loader
    gemm_wmma_f32<1, true><<<dim3(mb, H / TN), 256, 0, stream>>>(
        x, sums, u, batch, W3, b3, t0, M, F + H + 64, H);

    // 6) out = t0 @ W4 + b4          [M,256]x[256,128]
    gemm_wmma_f32<0, false><<<dim3(mb, F / TN), 256, 0, stream>>>(
        t0, nullptr, nullptr, nullptr, W4, b4, out, M, H, F);
}